// GraphNet_53730040873195
// MI455X (gfx1250) — compile-verified
//
#include <hip/hip_runtime.h>

#define NNODES  32768
#define BGRAPH  16
#define NPERG   2048
#define EEDGE   524288          // E (before self loops)
#define ETOT    557056          // E + N
#define EPERG   32768           // E / B
#define LNEPS   1e-5f
#define KCHUNK  64

typedef __attribute__((ext_vector_type(2))) float        v2f;
typedef __attribute__((ext_vector_type(8))) float        v8f;
typedef __attribute__((ext_vector_type(4))) unsigned int u32x4;
typedef __attribute__((ext_vector_type(4))) int          i32x4;
typedef __attribute__((ext_vector_type(8))) int          i32x8;

#ifndef __has_builtin
#define __has_builtin(x) 0
#endif
#if __has_builtin(__builtin_amdgcn_tensor_load_to_lds)
#define HAVE_TDM 1
#pragma message("CDNA5 probe: __builtin_amdgcn_tensor_load_to_lds AVAILABLE -> TDM staging path enabled")
#else
#define HAVE_TDM 0
#pragma message("CDNA5 probe: __builtin_amdgcn_tensor_load_to_lds NOT available -> cooperative LDS fallback")
#endif
#if defined(__clang_major__)
#if __clang_major__ >= 23
#pragma message("CDNA5 probe: clang >= 23 (amdgpu-toolchain lane, 6-arg TDM form)")
#else
#pragma message("CDNA5 probe: clang < 23 (ROCm lane, 5-arg TDM form)")
#endif
#endif

// ------------------------------------------------------------------ TDM D#
#if HAVE_TDM
// 2-D f32 tile load: rows [0,tile_d1) x cols [0,tile_d0) from a row-major
// tensor with line stride `stride0` (elements), into LDS at byte addr lds.
__device__ __forceinline__ void tdm_load_f32(unsigned lds, const float* gptr,
                                             unsigned tens_d0, unsigned tens_d1,
                                             unsigned stride0,
                                             unsigned tile_d0, unsigned tile_d1) {
    unsigned long long ga = (unsigned long long)(size_t)gptr;
    u32x4 g0 = { 1u,                                              // count=1 (valid)
                 lds,                                             // lds_addr
                 (unsigned)(ga & 0xFFFFFFFFu),                    // global_addr lo
                 (unsigned)((ga >> 32) & 0x01FFFFFFu) | (2u << 30) }; // ga hi | type=2
    i32x8 g1 = { (int)(2u << 16),                                 // data_size=4B, mask=0
                 (int)((tens_d0 & 0xFFFFu) << 16),                // dim0[15:0]
                 (int)((tens_d0 >> 16) | ((tens_d1 & 0xFFFFu) << 16)),
                 (int)((tens_d1 >> 16) | (tile_d0 << 16)),        // dim1 hi | tile_dim0
                 (int)tile_d1,                                    // tile_dim1 | tile_dim2=0
                 (int)stride0,                                    // dim0_stride lo32
                 0, 0 };                                          // stride hi, dim1_stride=0
    i32x4 z4 = { 0, 0, 0, 0 };
#if defined(__clang_major__) && __clang_major__ >= 23
    i32x8 z8 = { 0, 0, 0, 0, 0, 0, 0, 0 };
    __builtin_amdgcn_tensor_load_to_lds(g0, g1, z4, z4, z8, 0);
#else
    __builtin_amdgcn_tensor_load_to_lds(g0, g1, z4, z4, 0);
#endif
}
__device__ __forceinline__ void tdm_wait() {
#if __has_builtin(__builtin_amdgcn_s_wait_tensorcnt)
    __builtin_amdgcn_s_wait_tensorcnt(0);
#else
    asm volatile("s_wait_tensorcnt 0x0" ::: "memory");
#endif
}
#endif

// ---------------------------------------------------------------- reductions
__device__ __forceinline__ void blockReduce2(float& s, float& q) {
    __shared__ float shs[8], shq[8];
#pragma unroll
    for (int o = 16; o > 0; o >>= 1) { s += __shfl_down(s, o, 32); q += __shfl_down(q, o, 32); }
    int lane = threadIdx.x & 31, wid = threadIdx.x >> 5;
    if (lane == 0) { shs[wid] = s; shq[wid] = q; }
    __syncthreads();
    int nw = (blockDim.x + 31) >> 5;
    s = (threadIdx.x < (unsigned)nw) ? shs[threadIdx.x] : 0.f;
    q = (threadIdx.x < (unsigned)nw) ? shq[threadIdx.x] : 0.f;
    if (wid == 0) {
#pragma unroll
        for (int o = 16; o > 0; o >>= 1) { s += __shfl_down(s, o, 32); q += __shfl_down(q, o, 32); }
    }
}

// ------------------------------------------------------- coord LN (per graph)
__global__ void k_ln_coord(const float* __restrict__ x, const float* __restrict__ w,
                           const float* __restrict__ b, float* __restrict__ h0) {
    int g = blockIdx.x;
    const int per = NPERG * 3;
    const float* xg = x  + (size_t)g * per;
    float*       hg = h0 + (size_t)g * per;
    float s = 0.f, q = 0.f;
    for (int i = threadIdx.x; i < per; i += blockDim.x) { float v = xg[i]; s += v; q += v * v; }
    blockReduce2(s, q);
    __shared__ float st[2];
    if (threadIdx.x == 0) {
        float mean = s / (float)per;
        float var  = q / (float)per - mean * mean;
        st[0] = mean; st[1] = rsqrtf(var + LNEPS);
    }
    __syncthreads();
    float mean = st[0], inv = st[1];
    for (int i = threadIdx.x; i < per; i += blockDim.x) {
        int f = i % 3;
        hg[i] = (xg[i] - mean) * inv * w[f] + b[f];
    }
}

// --------------------------------------------- tiny dense layers (3/12/48 K)
__global__ void k_dense_small(const float* __restrict__ X, const float* __restrict__ W,
                              const float* __restrict__ bias, float* __restrict__ Y,
                              int K, int Nout, int total) {
    int i = blockIdx.x * blockDim.x + threadIdx.x;
    if (i >= total) return;
    int n = i / Nout, j = i - n * Nout;
    float acc = bias[j];
    const float* xr = X + (size_t)n * K;
    for (int k = 0; k < K; ++k) acc = fmaf(xr[k], W[(size_t)k * Nout + j], acc);
    Y[i] = fmaxf(acc, 0.f);
}

// ---------------- fp32 WMMA GEMM, register-blocked 16x64 per wave, LDS B ----
// grid.x = (M/128) * (Nout/64); block = 8 waves; B chunk staged by TDM.
__global__ void k_gemm_wmma_blk(const float* __restrict__ X, const float* __restrict__ W,
                                const float* __restrict__ bias, float* __restrict__ Y,
                                int M, int K, int Nout) {
    __shared__ float Bs[KCHUNK * 64];                 // 16 KB
    int nPanels = Nout >> 6;
    int pm = blockIdx.x / nPanels;
    int pn = blockIdx.x - pm * nPanels;
    int col0 = pn * 64;
    int wave = threadIdx.x >> 5;
    int lane = threadIdx.x & 31;
    int l  = lane & 15;
    int kk = (lane >> 4) << 1;                        // lanes 0-15: K 0,1 ; 16-31: K 2,3
    int tm = pm * 128 + wave * 16;
    const float* xrow = X + (size_t)(tm + l) * K;
    v8f c[4];
#pragma unroll
    for (int nt = 0; nt < 4; ++nt) c[nt] = (v8f){0.f,0.f,0.f,0.f,0.f,0.f,0.f,0.f};

    for (int k0 = 0; k0 < K; k0 += KCHUNK) {
#if HAVE_TDM
        if (wave == 0) {
            tdm_load_f32((unsigned)(size_t)&Bs[0], W + (size_t)k0 * Nout + col0,
                         (unsigned)Nout, (unsigned)K, (unsigned)Nout, 64u, (unsigned)KCHUNK);
            tdm_wait();
        }
#else
        for (int i = threadIdx.x; i < KCHUNK * 64; i += 256) {
            int kr = i >> 6, cc = i & 63;
            Bs[i] = W[(size_t)(k0 + kr) * Nout + col0 + cc];
        }
#endif
        __syncthreads();
        for (int k = 0; k < KCHUNK; k += 4) {
            v2f a;
            a.x = xrow[k0 + k + kk];
            a.y = xrow[k0 + k + kk + 1];
            const float* b0 = Bs + (k + kk) * 64 + l;
            const float* b1 = Bs + (k + kk + 1) * 64 + l;
#pragma unroll
            for (int nt = 0; nt < 4; ++nt) {
                v2f b;
                b.x = b0[nt * 16];
                b.y = b1[nt * 16];
                c[nt] = __builtin_amdgcn_wmma_f32_16x16x4_f32(false, a, false, b, (short)0, c[nt], false, false);
            }
        }
        __syncthreads();
    }
    int rbase = tm + ((lane >> 4) << 3);
#pragma unroll
    for (int nt = 0; nt < 4; ++nt) {
        int col = col0 + nt * 16 + l;
        float bv = bias[col];
#pragma unroll
        for (int r = 0; r < 8; ++r)
            Y[(size_t)(rbase + r) * Nout + col] = fmaxf(c[nt][r] + bv, 0.f);
    }
}

// ------------------------- h += atom_emb + aa_emb, accumulate LN stats/graph
__global__ void k_add_emb_stats(float* __restrict__ H, const int* __restrict__ atom_ids,
                                const int* __restrict__ aa_ids, const float* __restrict__ atom_emb,
                                const float* __restrict__ aa_emb, float* __restrict__ stats) {
    int idx = blockIdx.x * 1024 + threadIdx.x * 4;
    int n = idx / 768;
    int f = idx - n * 768;
    float4 hv = *(float4*)(H + (size_t)idx);
    int aid = atom_ids[n], rid = aa_ids[n];
    const float4 ae = *(const float4*)(atom_emb + (size_t)aid * 768 + f);
    const float4 re = *(const float4*)(aa_emb  + (size_t)rid * 768 + f);
    hv.x += ae.x + re.x; hv.y += ae.y + re.y; hv.z += ae.z + re.z; hv.w += ae.w + re.w;
    *(float4*)(H + (size_t)idx) = hv;
    float s = hv.x + hv.y + hv.z + hv.w;
    float q = hv.x * hv.x + hv.y * hv.y + hv.z * hv.z + hv.w * hv.w;
    blockReduce2(s, q);
    if (threadIdx.x == 0) {
        int g = blockIdx.x / ((NPERG * 768) / 1024);
        atomicAdd(&stats[2 * g],     s);
        atomicAdd(&stats[2 * g + 1], q);
    }
}

// --------------------------------------------------- node LN normalize (768)
__global__ void k_norm_node(float* __restrict__ H, const float* __restrict__ stats,
                            const float* __restrict__ w, const float* __restrict__ b) {
    int i   = blockIdx.x * blockDim.x + threadIdx.x;
    int idx = i * 4;
    int g   = idx / (NPERG * 768);
    int f   = idx % 768;
    const float denom = (float)(NPERG * 768);
    float mean = stats[2 * g] / denom;
    float inv  = rsqrtf(stats[2 * g + 1] / denom - mean * mean + LNEPS);
    float4 hv = *(float4*)(H + (size_t)idx);
    float4 wv = *(const float4*)(w + f);
    float4 bv = *(const float4*)(b + f);
    hv.x = (hv.x - mean) * inv * wv.x + bv.x;
    hv.y = (hv.y - mean) * inv * wv.y + bv.y;
    hv.z = (hv.z - mean) * inv * wv.z + bv.z;
    hv.w = (hv.w - mean) * inv * wv.w + bv.w;
    *(float4*)(H + (size_t)idx) = hv;
}

// -------------------- edge gather (he[row]+he[col])/2 + LDS-binned LN stats
__global__ void k_edge_gather(const float* __restrict__ he, const int* __restrict__ erow,
                              const int* __restrict__ ecol, float* __restrict__ EA,
                              float* __restrict__ stats) {
    __shared__ float shs[BGRAPH], shq[BGRAPH];
    if (threadIdx.x < BGRAPH) { shs[threadIdx.x] = 0.f; shq[threadIdx.x] = 0.f; }
    __syncthreads();
    long long t = (long long)blockIdx.x * blockDim.x + threadIdx.x;
    long long r = t >> 5;
    int f4 = (int)(t & 31) * 4;
    int a, bnode, g;
    if (r < EEDGE) { a = erow[r]; bnode = ecol[r]; g = (int)(r >> 15); }
    else           { int n = (int)(r - EEDGE); a = n; bnode = n; g = n >> 11; }
    float4 va = *(const float4*)(he + (size_t)a     * 128 + f4);
    float4 vb = *(const float4*)(he + (size_t)bnode * 128 + f4);
    float4 s;
    s.x = 0.5f * (va.x + vb.x); s.y = 0.5f * (va.y + vb.y);
    s.z = 0.5f * (va.z + vb.z); s.w = 0.5f * (va.w + vb.w);
    *(float4*)(EA + r * 128 + f4) = s;
    float ls = s.x + s.y + s.z + s.w;
    float lq = s.x * s.x + s.y * s.y + s.z * s.z + s.w * s.w;
    atomicAdd(&shs[g], ls);
    atomicAdd(&shq[g], lq);
    __syncthreads();
    if (threadIdx.x < BGRAPH) {
        atomicAdd(&stats[2 * threadIdx.x],     shs[threadIdx.x]);
        atomicAdd(&stats[2 * threadIdx.x + 1], shq[threadIdx.x]);
    }
}

// --------------------------------------------------- edge LN normalize (128)
__global__ void k_norm_edge(float* __restrict__ EA, const float* __restrict__ stats,
                            const float* __restrict__ w, const float* __restrict__ b) {
    long long i   = (long long)blockIdx.x * blockDim.x + threadIdx.x;
    long long idx = i * 4;
    long long r   = idx >> 7;
    int f = (int)(idx & 127);
    int g = (r < EEDGE) ? (int)(r >> 15) : (int)((r - EEDGE) >> 11);
    const float denom = (float)((EPERG + NPERG) * 128);
    float mean = stats[2 * g] / denom;
    float inv  = rsqrtf(stats[2 * g + 1] / denom - mean * mean + LNEPS);
    float4 hv = *(float4*)(EA + idx);
    float4 wv = *(const float4*)(w + f);
    float4 bv = *(const float4*)(b + f);
    hv.x = (hv.x - mean) * inv * wv.x + bv.x;
    hv.y = (hv.y - mean) * inv * wv.y + bv.y;
    hv.z = (hv.z - mean) * inv * wv.z + bv.z;
    hv.w = (hv.w - mean) * inv * wv.w + bv.w;
    *(float4*)(EA + idx) = hv;
}

// -------- glob GEMM + per-graph mean-pool; W_glob (64 KB) resident in LDS ---
// grid.x = 68 blocks; each wave handles one 1024-row chunk x all 128 columns,
// so edge_attr is streamed from HBM exactly once.
__global__ void k_gemm_pool(const float* __restrict__ EA, const float* __restrict__ Wg,
                            const float* __restrict__ bias, float* __restrict__ pooled) {
    __shared__ float Ws[128 * 128];                   // 64 KB (of 320 KB/WGP)
    int wave = threadIdx.x >> 5;
    int lane = threadIdx.x & 31;
#if HAVE_TDM
    if (wave == 0) {
        tdm_load_f32((unsigned)(size_t)&Ws[0], Wg, 128u, 128u, 128u, 128u, 128u);
        tdm_wait();
    }
#else
    for (int i = threadIdx.x; i < 128 * 128; i += 256) Ws[i] = Wg[i];
#endif
    __syncthreads();
    int chunk = blockIdx.x * 8 + wave;                // 544 chunks, graph-aligned
    int l  = lane & 15;
    int kk = (lane >> 4) << 1;
    float bvs[8], accs[8];
#pragma unroll
    for (int nt = 0; nt < 8; ++nt) { bvs[nt] = bias[nt * 16 + l]; accs[nt] = 0.f; }
    long long row0 = (long long)chunk * 1024;
    int g = (row0 < EEDGE) ? (int)(row0 >> 15) : (int)((row0 - EEDGE) >> 11);
    for (int t = 0; t < 64; ++t) {
        const float* xrow = EA + (row0 + t * 16 + l) * 128;
        v8f c[8];
#pragma unroll
        for (int nt = 0; nt < 8; ++nt) c[nt] = (v8f){0.f,0.f,0.f,0.f,0.f,0.f,0.f,0.f};
        for (int k = 0; k < 128; k += 4) {
            v2f a;
            a.x = xrow[k + kk];
            a.y = xrow[k + kk + 1];
            const float* b0 = Ws + (k + kk) * 128 + l;
            const float* b1 = Ws + (k + kk + 1) * 128 + l;
#pragma unroll
            for (int nt = 0; nt < 8; ++nt) {
                v2f b;
                b.x = b0[nt * 16];
                b.y = b1[nt * 16];
                c[nt] = __builtin_amdgcn_wmma_f32_16x16x4_f32(false, a, false, b, (short)0, c[nt], false, false);
            }
        }
#pragma unroll
        for (int nt = 0; nt < 8; ++nt)
#pragma unroll
            for (int r = 0; r < 8; ++r) accs[nt] += fmaxf(c[nt][r] + bvs[nt], 0.f);
    }
#pragma unroll
    for (int nt = 0; nt < 8; ++nt)
        atomicAdd(&pooled[g * 128 + nt * 16 + l], accs[nt]);
}

// ------------------------------------------- u = per-row LN of pooled means
__global__ void k_u(const float* __restrict__ pooled, const float* __restrict__ w,
                    const float* __restrict__ b, float* __restrict__ U) {
    int g = blockIdx.x, t = threadIdx.x;
    float p = pooled[g * 128 + t] * (1.f / (float)(EPERG + NPERG));
    float s = p, q = p * p;
    blockReduce2(s, q);
    __shared__ float st[2];
    if (threadIdx.x == 0) {
        float mean = s / 128.f;
        st[0] = mean; st[1] = rsqrtf(q / 128.f - mean * mean + LNEPS);
    }
    __syncthreads();
    U[g * 128 + t] = (p - st[0]) * st[1] * w[t] + b[t];
}

// ---------------------------------------------------------------------------
extern "C" void kernel_launch(void* const* d_in, const int* in_sizes, int n_in,
                              void* d_out, int out_size, void* d_ws, size_t ws_size,
                              hipStream_t stream) {
    const float* x         = (const float*)d_in[0];
    const int*   atom_ids  = (const int*)  d_in[1];
    const int*   aa_ids    = (const int*)  d_in[2];
    const int*   edge_idx  = (const int*)  d_in[3];
    const float* W_pre1 = (const float*)d_in[4],  *b_pre1 = (const float*)d_in[5];
    const float* W_pre2 = (const float*)d_in[6],  *b_pre2 = (const float*)d_in[7];
    const float* W_pre3 = (const float*)d_in[8],  *b_pre3 = (const float*)d_in[9];
    const float* W_pre4 = (const float*)d_in[10], *b_pre4 = (const float*)d_in[11];
    const float* W_dense = (const float*)d_in[12], *b_dense = (const float*)d_in[13];
    const float* atom_emb = (const float*)d_in[14], *aa_emb = (const float*)d_in[15];
    const float* w_ln_coord = (const float*)d_in[16], *b_ln_coord = (const float*)d_in[17];
    const float* w_ln_node  = (const float*)d_in[18], *b_ln_node  = (const float*)d_in[19];
    const float* W_edge = (const float*)d_in[20], *b_edge = (const float*)d_in[21];
    const float* w_ln_edge = (const float*)d_in[22], *b_ln_edge = (const float*)d_in[23];
    const float* W_glob = (const float*)d_in[24], *b_glob = (const float*)d_in[25];
    const float* w_ln_glob = (const float*)d_in[26], *b_ln_glob = (const float*)d_in[27];

    float* out = (float*)d_out;
    float* H  = out;                                   // [N,768]
    float* EA = out + (size_t)NNODES * 768;            // [E+N,128]
    float* U  = EA  + (size_t)ETOT   * 128;            // [B,128]

    float* ws = (float*)d_ws;
    float* h0 = ws;                                    // [N,3]
    float* h1 = h0 + (size_t)NNODES * 3;               // [N,12]
    float* h2 = h1 + (size_t)NNODES * 12;              // [N,48]
    float* h3 = h2 + (size_t)NNODES * 48;              // [N,192]
    float* h4 = h3 + (size_t)NNODES * 192;             // [N,768]
    float* he = h4 + (size_t)NNODES * 768;             // [N,128]
    float* nodeStats = he + (size_t)NNODES * 128;      // [B,2]
    float* edgeStats = nodeStats + 2 * BGRAPH;         // [B,2]
    float* pooled    = edgeStats + 2 * BGRAPH;         // [B,128]

    hipMemsetAsync(nodeStats, 0, (4 * BGRAPH + BGRAPH * 128) * sizeof(float), stream);

    // NodeInit
    k_ln_coord<<<BGRAPH, 256, 0, stream>>>(x, w_ln_coord, b_ln_coord, h0);
    int t1 = NNODES * 12;
    k_dense_small<<<(t1 + 255) / 256, 256, 0, stream>>>(h0, W_pre1, b_pre1, h1, 3, 12, t1);
    int t2 = NNODES * 48;
    k_dense_small<<<(t2 + 255) / 256, 256, 0, stream>>>(h1, W_pre2, b_pre2, h2, 12, 48, t2);
    int t3 = NNODES * 192;
    k_dense_small<<<(t3 + 255) / 256, 256, 0, stream>>>(h2, W_pre3, b_pre3, h3, 48, 192, t3);
    // 192 -> 768 (WMMA + TDM-staged B)
    k_gemm_wmma_blk<<<(NNODES / 128) * (768 / 64), 256, 0, stream>>>(h3, W_pre4, b_pre4, h4, NNODES, 192, 768);
    // 768 -> 768 (WMMA), result straight into output h slot
    k_gemm_wmma_blk<<<(NNODES / 128) * (768 / 64), 256, 0, stream>>>(h4, W_dense, b_dense, H, NNODES, 768, 768);
    k_add_emb_stats<<<(NNODES * 768) / 1024, 256, 0, stream>>>(H, atom_ids, aa_ids, atom_emb, aa_emb, nodeStats);
    k_norm_node<<<(NNODES * 768 / 4) / 256, 256, 0, stream>>>(H, nodeStats, w_ln_node, b_ln_node);

    // EdgeInit
    k_gemm_wmma_blk<<<(NNODES / 128) * (128 / 64), 256, 0, stream>>>(H, W_edge, b_edge, he, NNODES, 768, 128);
    k_edge_gather<<<(int)(((long long)ETOT * 32) / 256), 256, 0, stream>>>(he, edge_idx, edge_idx + EEDGE, EA, edgeStats);
    k_norm_edge<<<(int)(((long long)ETOT * 32) / 256), 256, 0, stream>>>(EA, edgeStats, w_ln_edge, b_ln_edge);

    // GlobalInit
    k_gemm_pool<<<(ETOT / 1024) / 8, 256, 0, stream>>>(EA, W_glob, b_glob, pooled);
    k_u<<<BGRAPH, 128, 0, stream>>>(pooled, w_ln_glob, b_ln_glob, U);
}